// QuantumKernelMethod_65481071403552
// MI455X (gfx1250) — compile-verified
//
#include <hip/hip_runtime.h>
#include <hip/hip_bf16.h>
#include <math.h>

// ---------------------------------------------------------------------------
// QuantumKernelMethod: K[m][n] = |<psi_x[m] | psi_y[n]>|^2
//   Stage 1: build statevectors (1024 complex amps) per batch row in LDS.
//   Stage 2: complex Gram matrix via fp32 WMMA, one pass over k in [0,1024):
//     Re  = xr.yr + xi.yi ;  Im = xi.yr - xr.yi  (3 accumulator banks)
// ---------------------------------------------------------------------------

#define NQ    10
#define DIM   1024         // 2^NQ
#define BATCH 4096
#define KDIM  2048         // [re | im]

typedef __attribute__((ext_vector_type(2))) float v2f;
typedef __attribute__((ext_vector_type(8))) float v8f;

// ---------------------------------------------------------------------------
// Stage 1: state preparation. One workgroup (128 threads) per batch element.
// mode 0 (A): out[b*2048 + i]   = re(i),  out[b*2048 + 1024 + i] = im(i)
// mode 1 (B): pair-packed K-major: element (k,col) ->
//             out[((k>>1)*ldn + col)*2 + (k&1)],   k = i (re) / 1024+i (im)
// ---------------------------------------------------------------------------
__global__ __launch_bounds__(128) void qkm_state_prep(
    const float* __restrict__ angles,   // BATCH x NQ
    const float* __restrict__ params,   // 2 x NQ x 3
    float* __restrict__ out,
    int mode, int ldn)
{
    __shared__ float sre[DIM];
    __shared__ float sim[DIM];

    const int tid = threadIdx.x;
    const int b   = blockIdx.x;

    // |0...0>
    for (int i = tid; i < DIM; i += 128) {
        sre[i] = (i == 0) ? 1.0f : 0.0f;
        sim[i] = 0.0f;
    }
    __syncthreads();

    // ---- encoding layer: RY(x[b][q]) on every qubit (real 2x2) ----
    for (int q = 0; q < NQ; ++q) {
        float c, s;
        sincosf(0.5f * angles[b * NQ + q], &s, &c);
        const int S = 512 >> q;                       // pair stride
        #pragma unroll
        for (int it = 0; it < 4; ++it) {              // 512 pairs / 128 threads
            int p  = tid + it * 128;
            int lt = p & (S - 1);
            int hi = p - lt;                          // l * S
            int i0 = (hi << 1) + lt;
            int i1 = i0 + S;
            float r0 = sre[i0], m0 = sim[i0];
            float r1 = sre[i1], m1 = sim[i1];
            sre[i0] = c * r0 - s * r1;  sim[i0] = c * m0 - s * m1;
            sre[i1] = s * r0 + c * r1;  sim[i1] = s * m0 + c * m1;
        }
        __syncthreads();
    }

    // ---- variational layers: Rot + CNOT ladder ----
    for (int layer = 0; layer < 2; ++layer) {
        for (int q = 0; q < NQ; ++q) {
            const float* p3 = params + ((layer * NQ) + q) * 3;
            float phi = p3[0], theta = p3[1], omega = p3[2];
            float c, s, ca, sa, cb, sb;
            sincosf(0.5f * theta, &s, &c);
            sincosf(0.5f * (phi + omega), &sa, &ca);  // ep = (ca, -sa)
            sincosf(0.5f * (phi - omega), &sb, &cb);  // em = (cb, -sb)
            // Rot = [[ep*c, -conj(em)*s], [em*s, conj(ep)*c]]
            float g00r =  c * ca, g00i = -c * sa;
            float g01r = -s * cb, g01i = -s * sb;
            float g10r =  s * cb, g10i = -s * sb;
            float g11r =  c * ca, g11i =  c * sa;

            const int S = 512 >> q;
            #pragma unroll
            for (int it = 0; it < 4; ++it) {
                int p  = tid + it * 128;
                int lt = p & (S - 1);
                int hi = p - lt;
                int i0 = (hi << 1) + lt;
                int i1 = i0 + S;
                float r0 = sre[i0], m0 = sim[i0];
                float r1 = sre[i1], m1 = sim[i1];
                sre[i0] = g00r*r0 - g00i*m0 + g01r*r1 - g01i*m1;
                sim[i0] = g00r*m0 + g00i*r0 + g01r*m1 + g01i*r1;
                sre[i1] = g10r*r0 - g10i*m0 + g11r*r1 - g11i*m1;
                sim[i1] = g10r*m0 + g10i*r0 + g11r*m1 + g11i*r1;
            }
            __syncthreads();
        }
        for (int q = 0; q < NQ - 1; ++q) {            // CNOT(q, q+1)
            const int T = 256 >> q;                   // target stride
            #pragma unroll
            for (int it = 0; it < 2; ++it) {          // 256 swaps / 128 threads
                int p  = tid + it * 128;
                int lt = p & (T - 1);
                int hi = p - lt;                      // l * T
                int i0 = (hi << 2) + 2 * T + lt;      // ctrl=1, tgt=0
                int i1 = i0 + T;
                float r0 = sre[i0], m0 = sim[i0];
                sre[i0] = sre[i1]; sim[i0] = sim[i1];
                sre[i1] = r0;      sim[i1] = m0;
            }
            __syncthreads();
        }
    }

    // ---- write out ----
    if (mode == 0) {
        float* row = out + (size_t)b * KDIM;
        for (int i = tid; i < DIM; i += 128) {
            row[i]       = sre[i];
            row[i + DIM] = sim[i];
        }
    } else {
        for (int i = tid; i < DIM; i += 128) {
            const int slot = i & 1;
            const size_t pre = (size_t)(i >> 1) * ldn;          // k = i
            const size_t pim = (size_t)(512 + (i >> 1)) * ldn;  // k = 1024+i
            out[(pre + b) * 2 + slot] = sre[i];
            out[(pim + b) * 2 + slot] = sim[i];
        }
    }
}

// ---------------------------------------------------------------------------
// Stage 2: K = Re^2 + Im^2 via fp32 WMMA 16x16x4.
//   A : 4096 x 2048 row-major  ([xr | xi] per row)
//   Bp: pair-packed K-major, pair P=k>>1: Bp[(P*4096 + col)*2 + (k&1)]
//       P in [0,512)   -> yr rows,  P in [512,1024) -> yi rows
// Per k-step (K=4 per phase, both phases fused => K=8 of work / 16 WMMAs):
//   accR  += aLo*bLo + aHi*bHi
//   accIp += aHi*bLo ;  accIn += aLo*bHi ;  Im = accIp - accIn
// Each wave: 32x32 tile. Block: 8 waves -> 128x64 tile of K.
// ---------------------------------------------------------------------------
__global__ __launch_bounds__(256) void qkm_gram(
    const float* __restrict__ A,
    const float* __restrict__ Bp,
    float* __restrict__ Kout)
{
    const int N = BATCH, LDA = KDIM;

    const int lane = threadIdx.x & 31;
    const int wave = threadIdx.x >> 5;
    const int lh   = lane >> 4;        // lane half: K pair {0,1} vs {2,3}
    const int lm   = lane & 15;        // row (A) / col (B) within fragment
    const int wm   = wave & 3;
    const int wn   = wave >> 2;

    const int m_base = blockIdx.x * 128 + wm * 32;
    const int n_base = blockIdx.y * 64  + wn * 32;

    const float* Arow0 = A + (size_t)(m_base + lm)      * LDA + 2 * lh;
    const float* Arow1 = A + (size_t)(m_base + 16 + lm) * LDA + 2 * lh;
    // B pointers: pair index P = (k>>1) + lh ; +512 pairs for the yi half
    const float* B0 = Bp + ((size_t)lh        * BATCH + n_base + lm) * 2;
    const float* B1 = Bp + ((size_t)(lh + 512) * BATCH + n_base + lm) * 2;

    v8f accR[2][2]  = {};
    v8f accIp[2][2] = {};
    v8f accIn[2][2] = {};

    #pragma unroll 2
    for (int k = 0; k < DIM; k += 4) {
        v2f aLo[2], aHi[2], bLo[2], bHi[2];
        aLo[0] = *(const v2f*)(Arow0 + k);
        aLo[1] = *(const v2f*)(Arow1 + k);
        aHi[0] = *(const v2f*)(Arow0 + DIM + k);
        aHi[1] = *(const v2f*)(Arow1 + DIM + k);

        const size_t pofs = (size_t)(k >> 1) * (2 * BATCH);  // 2 pairs / step
        bLo[0] = *(const v2f*)(B0 + pofs);
        bLo[1] = *(const v2f*)(B0 + pofs + 32);
        bHi[0] = *(const v2f*)(B1 + pofs);
        bHi[1] = *(const v2f*)(B1 + pofs + 32);

        #pragma unroll
        for (int mi = 0; mi < 2; ++mi)
            #pragma unroll
            for (int ni = 0; ni < 2; ++ni) {
                accR[mi][ni] = __builtin_amdgcn_wmma_f32_16x16x4_f32(
                    false, aLo[mi], false, bLo[ni], (short)0, accR[mi][ni],
                    false, false);
                accR[mi][ni] = __builtin_amdgcn_wmma_f32_16x16x4_f32(
                    false, aHi[mi], false, bHi[ni], (short)0, accR[mi][ni],
                    false, false);
                accIp[mi][ni] = __builtin_amdgcn_wmma_f32_16x16x4_f32(
                    false, aHi[mi], false, bLo[ni], (short)0, accIp[mi][ni],
                    false, false);
                accIn[mi][ni] = __builtin_amdgcn_wmma_f32_16x16x4_f32(
                    false, aLo[mi], false, bHi[ni], (short)0, accIn[mi][ni],
                    false, false);
            }
    }

    // C/D layout: VGPR v, lane half lh -> row = v + 8*lh; col = lm
    #pragma unroll
    for (int mi = 0; mi < 2; ++mi) {
        #pragma unroll
        for (int ni = 0; ni < 2; ++ni) {
            const int col = n_base + ni * 16 + lm;
            #pragma unroll
            for (int v = 0; v < 8; ++v) {
                const int row = m_base + mi * 16 + v + 8 * lh;
                const float re = accR[mi][ni][v];
                const float im = accIp[mi][ni][v] - accIn[mi][ni][v];
                __builtin_nontemporal_store(re * re + im * im,
                                            &Kout[(size_t)row * N + col]);
            }
        }
    }
}

// ---------------------------------------------------------------------------
extern "C" void kernel_launch(void* const* d_in, const int* in_sizes, int n_in,
                              void* d_out, int out_size, void* d_ws, size_t ws_size,
                              hipStream_t stream) {
    (void)in_sizes; (void)n_in; (void)out_size; (void)ws_size;
    const float* x      = (const float*)d_in[0];   // 4096 x 10
    const float* y      = (const float*)d_in[1];   // 4096 x 10
    const float* params = (const float*)d_in[2];   // 2 x 10 x 3

    float* Ax = (float*)d_ws;                          // 4096*2048 f32 (32 MB)
    float* By = Ax + (size_t)BATCH * KDIM;             // 2048*4096 f32 (32 MB)
    float* K  = (float*)d_out;                         // 4096*4096 f32

    qkm_state_prep<<<BATCH, 128, 0, stream>>>(x, params, Ax, 0, KDIM);
    qkm_state_prep<<<BATCH, 128, 0, stream>>>(y, params, By, 1, BATCH);

    dim3 grid(BATCH / 128, BATCH / 64);
    qkm_gram<<<grid, 256, 0, stream>>>(Ax, By, K);
}